// GCNOneLayer_59554016526437
// MI455X (gfx1250) — compile-verified
//
#include <hip/hip_runtime.h>
#include <math.h>

// ---------------------------------------------------------------------------
// GCN one layer on MI455X (gfx1250).
// Dominant cost: h = x @ W with x = 1.55 GB fp32  ->  HBM-bound (~67us at
// 23.3 TB/s). 16-row tiles with V_WMMA_F32_16X16X4_F32 (fp32 WMMA, same
// precision class as reference), split-K across 64 slices/tile, fp32 global
// atomics for partial-tile reduction and graph scatter.
//
// Structure notes:
//  * Inner loop is branchless: out-of-range A rows / B cols are CLAMPED (not
//    masked) — those accumulator rows/cols are garbage but never stored.
//  * WMMA is convergent -> LLVM cannot runtime-unroll; so the K loop is
//    chunks of a compile-time-8 fully-unrolled body + a tiny rolled tail.
//  * Loop bounds go through readfirstlane so loop control is scalar
//    (s_cbranch, EXEC stays all-1s as WMMA requires).
// ---------------------------------------------------------------------------

constexpr int N   = 2708;
constexpr int E   = 10556;
constexpr int C   = 143300;      // 1433 * 100
constexpr int OUT = 7;

constexpr int NT     = (N + 15) / 16;        // 170 row tiles
constexpr int KQ     = C / 4;                // 35825 K-quads (C % 4 == 0)
constexpr int SKY    = 8;                    // split-K blocks per tile (grid.y)
constexpr int WAVES  = 8;                    // waves per block (256 threads)
constexpr int SLICES = SKY * WAVES;          // 64 K-slices per tile
constexpr int QPS    = (KQ + SLICES - 1) / SLICES;  // 560 quads per slice
constexpr int CHUNK  = 8;                    // fully-unrolled WMMAs per chunk

typedef float v2f __attribute__((ext_vector_type(2)));
typedef float v8f __attribute__((ext_vector_type(8)));

__device__ inline void atomAddF(float* p, float v) {
    __hip_atomic_fetch_add(p, v, __ATOMIC_RELAXED, __HIP_MEMORY_SCOPE_AGENT);
}

// -------------------------------- setup ------------------------------------

__global__ void zero_kernel(float* p, int n) {
    int t = blockIdx.x * blockDim.x + threadIdx.x;
    if (t < n) p[t] = 0.0f;
}

__global__ void deg_kernel(const long long* __restrict__ ei, float* __restrict__ deg) {
    int e = blockIdx.x * blockDim.x + threadIdx.x;
    if (e < E) {
        int d = (int)ei[E + e];          // edge_index[1][e]
        atomAddF(&deg[d], 1.0f);
    }
}

__global__ void dinv_kernel(const float* __restrict__ deg, float* __restrict__ dinv) {
    int i = blockIdx.x * blockDim.x + threadIdx.x;
    if (i < N) dinv[i] = rsqrtf(deg[i] + 1.0f);   // +1 = self loop; always > 0
}

// ------------------------- skinny GEMM: h = x @ W --------------------------
// A tile: x[row16, k:k+4] (16x4 f32), B tile: W[k:k+4, 0:16].
// ISA layouts (CDNA5 ISA):
//   A 16x4 f32 : lane L holds row (L&15); VGPRs = K {2*(L>>4), 2*(L>>4)+1}
//   B 4x16 f32 : lane L holds col (L&15); VGPRs = K {2*(L>>4), 2*(L>>4)+1}
//   C/D 16x16  : VGPR v, lane L -> row v + 8*(L>>4), col (L&15)
// Lanes L and L+16 together read a dense 16B per row per step -> perfectly
// dense streaming of x. W (4 MB) stays L2-resident.

__global__ __launch_bounds__(256) void gcn_gemm_wmma(
    const float* __restrict__ x, const float* __restrict__ W,
    float* __restrict__ h)
{
    const int lane = threadIdx.x & 31;
    const int wave = threadIdx.x >> 5;
    const int n    = lane & 15;
    const int hi   = lane >> 4;

    const int rowBase = blockIdx.x * 16;
    int rowA = rowBase + n;                  // A-matrix row for this lane
    if (rowA > N - 1) rowA = N - 1;          // clamp: garbage rows never stored
    int nb = (n < OUT) ? n : (OUT - 1);      // clamp: garbage cols never stored

    // Wave-uniform K-slice bounds, pinned to SGPRs for scalar loop control.
    const int slice = blockIdx.y * WAVES + wave;
    int kb0 = __builtin_amdgcn_readfirstlane(slice * QPS);
    int kb1 = slice * QPS + QPS;
    if (kb1 > KQ) kb1 = KQ;
    kb1 = __builtin_amdgcn_readfirstlane(kb1);

    // Strength-reduced streaming pointers (8B-aligned A, L2-resident B).
    const float* ap = x + (size_t)rowA * (size_t)C + (size_t)(kb0 << 2) + (hi << 1);
    const float* bp = W + ((size_t)(kb0 << 2) + (hi << 1)) * OUT + nb;

    v8f acc = {0.f, 0.f, 0.f, 0.f, 0.f, 0.f, 0.f, 0.f};

    const int len = kb1 - kb0;               // 560 (or 545 for last slice)
    const int nCh = len / CHUNK;

    for (int c = 0; c < nCh; ++c) {          // scalar-bounded outer loop
        #pragma unroll
        for (int j = 0; j < CHUNK; ++j) {    // constant trip -> full unroll
            v2f a = *(const v2f*)(ap + 4 * j);           // global_load_b64
            v2f bb;
            bb.x = bp[4 * OUT * j];                      // W[k][nb]
            bb.y = bp[4 * OUT * j + OUT];                // W[k+1][nb]
            acc = __builtin_amdgcn_wmma_f32_16x16x4_f32(
                false, a, false, bb, (short)0, acc, false, false);
        }
        ap += 4 * CHUNK;
        bp += 4 * OUT * CHUNK;
    }
    for (int r = nCh * CHUNK; r < len; ++r) {  // <=7 trips, last slice only
        v2f a = *(const v2f*)ap;
        v2f bb;
        bb.x = bp[0];
        bb.y = bp[OUT];
        acc = __builtin_amdgcn_wmma_f32_16x16x4_f32(
            false, a, false, bb, (short)0, acc, false, false);
        ap += 4;
        bp += 4 * OUT;
    }

    // Reduce split-K partials into h[N][OUT] with native fp32 atomics.
    if (n < OUT) {
        float* hp = h + (size_t)(rowBase + hi * 8) * OUT + n;
        if (rowBase + 16 <= N) {             // uniform: full tile, no guards
            #pragma unroll
            for (int v = 0; v < 8; ++v) atomAddF(hp + v * OUT, acc[v]);
        } else {                             // ragged last tile only
            #pragma unroll
            for (int v = 0; v < 8; ++v) {
                int r = rowBase + hi * 8 + v;
                if (r < N) atomAddF(hp + v * OUT, acc[v]);
            }
        }
    }
}

// --------------------------- message passing -------------------------------

__global__ void logits_init_kernel(const float* __restrict__ h,
                                   const float* __restrict__ dinv,
                                   const float* __restrict__ b,
                                   float* __restrict__ logits)
{
    int t = blockIdx.x * blockDim.x + threadIdx.x;
    if (t < N * OUT) {
        int i = t / OUT;
        int j = t - i * OUT;
        float di = dinv[i];
        logits[t] = h[t] * di * di + b[j];      // self-loop term + bias
    }
}

__global__ void edge_scatter_kernel(const long long* __restrict__ ei,
                                    const float* __restrict__ h,
                                    const float* __restrict__ dinv,
                                    float* __restrict__ logits)
{
    int e = blockIdx.x * blockDim.x + threadIdx.x;
    if (e < E) {
        int s = (int)ei[e];
        int d = (int)ei[E + e];
        float nm = dinv[s] * dinv[d];
        #pragma unroll
        for (int j = 0; j < OUT; ++j)
            atomAddF(&logits[d * OUT + j], h[s * OUT + j] * nm);
    }
}

__global__ void logsoftmax_kernel(const float* __restrict__ logits,
                                  float* __restrict__ out)
{
    int i = blockIdx.x * blockDim.x + threadIdx.x;
    if (i < N) {
        float l[OUT];
        float m = -INFINITY;
        #pragma unroll
        for (int j = 0; j < OUT; ++j) { l[j] = logits[i * OUT + j]; m = fmaxf(m, l[j]); }
        float s = 0.f;
        #pragma unroll
        for (int j = 0; j < OUT; ++j) s += __expf(l[j] - m);
        float lse = m + logf(s);
        #pragma unroll
        for (int j = 0; j < OUT; ++j) out[i * OUT + j] = l[j] - lse;
    }
}

// -------------------------------- launch -----------------------------------

extern "C" void kernel_launch(void* const* d_in, const int* in_sizes, int n_in,
                              void* d_out, int out_size, void* d_ws, size_t ws_size,
                              hipStream_t stream)
{
    const float*     x  = (const float*)d_in[0];       // [N, C]
    const float*     W  = (const float*)d_in[1];       // [C, OUT]
    const float*     b  = (const float*)d_in[2];       // [OUT]
    const long long* ei = (const long long*)d_in[3];   // [2, E] int64

    // workspace layout (floats): h[N*7] | deg[N] | dinv[N] | logits[N*7]
    float* ws     = (float*)d_ws;
    float* h      = ws;
    float* deg    = ws + N * 7;
    float* dinv   = ws + N * 8;
    float* logits = ws + N * 9;

    const int B = 256;
    zero_kernel<<<(N * 8 + B - 1) / B, B, 0, stream>>>(ws, N * 8);   // h + deg
    deg_kernel<<<(E + B - 1) / B, B, 0, stream>>>(ei, deg);
    dinv_kernel<<<(N + B - 1) / B, B, 0, stream>>>(deg, dinv);

    gcn_gemm_wmma<<<dim3(NT, SKY), B, 0, stream>>>(x, W, h);

    logits_init_kernel<<<(N * OUT + B - 1) / B, B, 0, stream>>>(h, dinv, b, logits);
    edge_scatter_kernel<<<(E + B - 1) / B, B, 0, stream>>>(ei, h, dinv, logits);
    logsoftmax_kernel<<<(N + B - 1) / B, B, 0, stream>>>(logits, (float*)d_out);
}